// Head_29420525977708
// MI455X (gfx1250) — compile-verified
//
#include <hip/hip_runtime.h>
#include <hip/hip_bf16.h>

// ---------------------------------------------------------------------------
// Head attention, algebraically collapsed for D_IN=2.
//   logits[b,c,e] = yhat_c^T G yhat_e   (G = [Wq;bq]^T [Wk;bk], 3x3)
//   out[b,c,:]    = s0*u0 + s1*u1 + ub  (u_i = Wv_i @ Wo, ub = bv@Wo + bo)
// where (s0,s1) come from a causal softmax over the rank-3 logits.
// The 16x16 logit tiles are computed with V_WMMA_F32_16X16X4_F32.
// ---------------------------------------------------------------------------

typedef __attribute__((ext_vector_type(2))) float v2f;
typedef __attribute__((ext_vector_type(8))) float v8f;

#define BB     16
#define CC     2048
#define NCB    128       // CC / 16 c-blocks
#define ECHUNK 512
#define NCHUNK 4         // CC / ECHUNK
#define NEG_HUGE (-3.0e38f)

#if defined(__has_builtin)
#if __has_builtin(__builtin_amdgcn_wmma_f32_16x16x4_f32)
#define HAS_WMMA4 1
#endif
#endif
#ifndef HAS_WMMA4
#define HAS_WMMA4 0
#endif

// ---- Kernel 0: G[3][3] = [Wq0;Wq1;bq] . [Wk0;Wk1;bk] (dot over 2048) -------
__global__ __launch_bounds__(256) void gram_kernel(
    const float* __restrict__ Wq, const float* __restrict__ bq,
    const float* __restrict__ Wk, const float* __restrict__ bk,
    float* __restrict__ G) {
  float acc[9];
#pragma unroll
  for (int j = 0; j < 9; ++j) acc[j] = 0.0f;
  for (int h = threadIdx.x; h < CC; h += 256) {
    float q0 = Wq[h], q1 = Wq[CC + h], q2 = bq[h];
    float k0 = Wk[h], k1 = Wk[CC + h], k2 = bk[h];
    acc[0] += q0 * k0; acc[1] += q0 * k1; acc[2] += q0 * k2;
    acc[3] += q1 * k0; acc[4] += q1 * k1; acc[5] += q1 * k2;
    acc[6] += q2 * k0; acc[7] += q2 * k1; acc[8] += q2 * k2;
  }
  // wave32 reduce
#pragma unroll
  for (int j = 0; j < 9; ++j)
#pragma unroll
    for (int off = 16; off >= 1; off >>= 1)
      acc[j] += __shfl_xor(acc[j], off, 32);
  __shared__ float sm[8][9];
  int wid = threadIdx.x >> 5, lane = threadIdx.x & 31;
  if (lane == 0)
#pragma unroll
    for (int j = 0; j < 9; ++j) sm[wid][j] = acc[j];
  __syncthreads();
  if (threadIdx.x == 0) {
#pragma unroll
    for (int j = 0; j < 9; ++j) {
      float s = 0.0f;
      for (int w = 0; w < 8; ++w) s += sm[w][j];
      G[j] = s;
    }
  }
}

// ---- Kernel 1: U = {u0, u1, ub}: u_i[d] = sum_h Wv[i,h]*Wo[h,d], ub += bo --
__global__ __launch_bounds__(256) void proj_kernel(
    const float* __restrict__ Wv, const float* __restrict__ bv,
    const float* __restrict__ Wo, const float* __restrict__ bo,
    float* __restrict__ U) {
  int d = blockIdx.x * 256 + threadIdx.x;  // 0..2047
  float t0 = 0.0f, t1 = 0.0f, t2 = 0.0f;
  for (int h = 0; h < CC; ++h) {
    float w = Wo[(size_t)h * CC + d];  // coalesced row of Wo
    t0 = fmaf(Wv[h], w, t0);
    t1 = fmaf(Wv[CC + h], w, t1);
    t2 = fmaf(bv[h], w, t2);
  }
  U[d] = t0;
  U[CC + d] = t1;
  U[2 * CC + d] = t2 + bo[d];
}

// ---- Kernel 2: causal softmax collapse via WMMA f32 16x16x4 ----------------
// One wave per (batch, 16-row c-block, e-chunk). Emits per-row partial
// (m, S, A0, A1) in shifted-softmax form so chunks combine exactly.
__global__ __launch_bounds__(32) void attn_kernel(
    const float* __restrict__ x,   // [B, C, 2]
    const float* __restrict__ G,   // 3x3 row-major
    float* __restrict__ P) {       // [B][NCB][16][NCHUNK][4]
  const int b = blockIdx.x, cblk = blockIdx.y, chk = blockIdx.z;
  const int lane = threadIdx.x;
  const int m16 = lane & 15;
  const bool hi = lane >= 16;
  const int cBase = cblk * 16;
  const int eStart = chk * ECHUNK;

  // Uniform 3x3 Gram (scalar loads).
  float g[9];
#pragma unroll
  for (int j = 0; j < 9; ++j) g[j] = G[j];

  // Row data for A-matrix: row M = m16, z_j = x0*G0j + x1*G1j + G2j.
  const int c_m = cBase + m16;
  const float xq0 = x[((size_t)b * CC + c_m) * 2 + 0];
  const float xq1 = x[((size_t)b * CC + c_m) * 2 + 1];
  const float z0 = fmaf(xq0, g[0], fmaf(xq1, g[3], g[6]));
  const float z1 = fmaf(xq0, g[1], fmaf(xq1, g[4], g[7]));
  const float z2 = fmaf(xq0, g[2], fmaf(xq1, g[5], g[8]));
#if HAS_WMMA4
  // 16x4 f32 A layout: lanes 0-15 hold K={0,1}, lanes 16-31 hold K={2,3}.
  v2f Amat;
  Amat.x = hi ? z2 : z0;
  Amat.y = hi ? 0.0f : z1;
#endif

  // Per-lane online-softmax state for 8 row slots (rows cBase + r + 8*hi).
  float mx[8], S[8], A0[8], A1[8];
#pragma unroll
  for (int r = 0; r < 8; ++r) { mx[r] = NEG_HUGE; S[r] = 0.0f; A0[r] = 0.0f; A1[r] = 0.0f; }

  const int lim = cBase + 16;
  const int eEnd = (eStart + ECHUNK < lim) ? (eStart + ECHUNK) : lim;

  for (int eBase = eStart; eBase < eEnd; eBase += 16) {
    const int e = eBase + m16;
    const float xe0 = x[((size_t)b * CC + e) * 2 + 0];
    const float xe1 = x[((size_t)b * CC + e) * 2 + 1];
    float lg[8];
#if HAS_WMMA4
    // 4x16 f32 B assumed mirror layout: lanes = N, halves = K pairs.
    v2f Bmat;
    Bmat.x = hi ? 1.0f : xe0;
    Bmat.y = hi ? 0.0f : xe1;
    v8f Cz = {};
    v8f D = __builtin_amdgcn_wmma_f32_16x16x4_f32(
        false, Amat, false, Bmat, (short)0, Cz, false, false);
#pragma unroll
    for (int r = 0; r < 8; ++r) lg[r] = D[r];
#else
    // Fallback: fetch z of the target row via cross-lane shuffle.
#pragma unroll
    for (int r = 0; r < 8; ++r) {
      int src = r + (hi ? 8 : 0);
      float zz0 = __shfl(z0, src, 32);
      float zz1 = __shfl(z1, src, 32);
      float zz2 = __shfl(z2, src, 32);
      lg[r] = fmaf(zz0, xe0, fmaf(zz1, xe1, zz2));
    }
#endif
#pragma unroll
    for (int r = 0; r < 8; ++r) {
      const int row = cBase + r + (hi ? 8 : 0);
      const bool ok = (e <= row);                 // causal mask
      const float l = ok ? lg[r] : NEG_HUGE;
      const float mn = fmaxf(mx[r], l);
      const float corr = __expf(mx[r] - mn);
      const float p = ok ? __expf(l - mn) : 0.0f;
      S[r] = fmaf(S[r], corr, p);
      A0[r] = fmaf(A0[r], corr, p * xe0);
      A1[r] = fmaf(A1[r], corr, p * xe1);
      mx[r] = mn;
    }
  }

  // Merge the 16 lanes of each half (xor masks 1..8 stay within a half).
#pragma unroll
  for (int r = 0; r < 8; ++r) {
    float mr = mx[r];
#pragma unroll
    for (int off = 8; off >= 1; off >>= 1)
      mr = fmaxf(mr, __shfl_xor(mr, off, 32));
    const float sc = __expf(mx[r] - mr);
    float s = S[r] * sc, a0 = A0[r] * sc, a1 = A1[r] * sc;
#pragma unroll
    for (int off = 8; off >= 1; off >>= 1) {
      s += __shfl_xor(s, off, 32);
      a0 += __shfl_xor(a0, off, 32);
      a1 += __shfl_xor(a1, off, 32);
    }
    mx[r] = mr; S[r] = s; A0[r] = a0; A1[r] = a1;
  }
  if (m16 == 0) {
#pragma unroll
    for (int r = 0; r < 8; ++r) {
      const int row = r + (hi ? 8 : 0);
      float* p = P + ((((size_t)b * NCB + cblk) * 16 + row) * NCHUNK + chk) * 4;
      p[0] = mx[r]; p[1] = S[r]; p[2] = A0[r]; p[3] = A1[r];
    }
  }
}

// ---- Kernel 3: combine chunk partials, stream 268 MB output ----------------
__global__ __launch_bounds__(256) void out_kernel(
    const float* __restrict__ P, const float* __restrict__ U,
    float* __restrict__ out) {
  const int row = blockIdx.x;          // 0 .. B*C-1
  const int b = row >> 11, c = row & (CC - 1);
  const int cblk = c >> 4, rr = c & 15;
  const float* p = P + ((((size_t)b * NCB + cblk) * 16 + rr) * NCHUNK) * 4;
  float mt = NEG_HUGE;
#pragma unroll
  for (int k = 0; k < NCHUNK; ++k) mt = fmaxf(mt, p[k * 4 + 0]);
  float S = 0.0f, A0 = 0.0f, A1 = 0.0f;
#pragma unroll
  for (int k = 0; k < NCHUNK; ++k) {
    const float sc = __expf(p[k * 4 + 0] - mt);
    S = fmaf(p[k * 4 + 1], sc, S);
    A0 = fmaf(p[k * 4 + 2], sc, A0);
    A1 = fmaf(p[k * 4 + 3], sc, A1);
  }
  const float inv = 1.0f / S;
  const float s0 = A0 * inv, s1 = A1 * inv;
  const float4* u0 = (const float4*)(U);
  const float4* u1 = (const float4*)(U + CC);
  const float4* ub = (const float4*)(U + 2 * CC);
  float4* o = (float4*)(out + (size_t)row * CC);
#pragma unroll
  for (int k = 0; k < 2; ++k) {
    const int i = threadIdx.x + k * 256;   // 512 float4 per row
    const float4 a = u0[i], bb = u1[i], cc = ub[i];
    float4 r;
    r.x = fmaf(s0, a.x, fmaf(s1, bb.x, cc.x));
    r.y = fmaf(s0, a.y, fmaf(s1, bb.y, cc.y));
    r.z = fmaf(s0, a.z, fmaf(s1, bb.z, cc.z));
    r.w = fmaf(s0, a.w, fmaf(s1, bb.w, cc.w));
    o[i] = r;
  }
}

extern "C" void kernel_launch(void* const* d_in, const int* in_sizes, int n_in,
                              void* d_out, int out_size, void* d_ws, size_t ws_size,
                              hipStream_t stream) {
  const float* x  = (const float*)d_in[0];
  const float* Wk = (const float*)d_in[1];
  const float* bk = (const float*)d_in[2];
  const float* Wq = (const float*)d_in[3];
  const float* bq = (const float*)d_in[4];
  const float* Wv = (const float*)d_in[5];
  const float* bv = (const float*)d_in[6];
  const float* Wo = (const float*)d_in[7];
  const float* bo = (const float*)d_in[8];
  float* out = (float*)d_out;

  // Workspace layout (floats): G[16] | U[3*2048] | partials[B*NCB*16*NCHUNK*4]
  float* ws = (float*)d_ws;
  float* G = ws;
  float* U = ws + 16;
  float* P = ws + 16 + 3 * CC;   // ~2 MB total, well under typical ws_size

  gram_kernel<<<1, 256, 0, stream>>>(Wq, bq, Wk, bk, G);
  proj_kernel<<<CC / 256, 256, 0, stream>>>(Wv, bv, Wo, bo, U);
  dim3 g2(BB, NCB, NCHUNK);
  attn_kernel<<<g2, 32, 0, stream>>>(x, G, P);
  out_kernel<<<BB * CC, 256, 0, stream>>>(P, U, out);
}